// MLP_APPNP_80676665688564
// MI455X (gfx1250) — compile-verified
//
#include <hip/hip_runtime.h>

// MLP + APPNP(K=10, alpha=0.1) for MI455X (gfx1250, wave32).
// GEMMs use V_WMMA_F32_16X16X4_F32 (fp32-exact vs reference).
// W1 k-panel staged in LDS *transposed* (pad-34 stride) so each B fragment is a
// single aligned ds_load_b64 (no register repacking before v_wmma).
// Propagation uses hardware global_atomic_add_f32 scatter; working set is L2-resident.

typedef float v2f __attribute__((ext_vector_type(2)));
typedef float v8f __attribute__((ext_vector_type(8)));

#define IN_F  512
#define HID_F 256
#define OUT_F 64
#define ALPHA_F 0.1f
#define ONE_MINUS_ALPHA 0.9f
#define W1P_STRIDE 34   // 32 k-values + pad; even => b64-aligned, conflict-free banks

// ---------------- degree / norm precompute ----------------

__global__ void k_init_deg(float* deg, long long n) {
    long long i = (long long)blockIdx.x * blockDim.x + threadIdx.x;
    if (i < n) deg[i] = 1.0f;   // self-loop
}

__global__ void k_deg_accum(const long long* __restrict__ dst, float* deg, long long e_cnt) {
    long long e = (long long)blockIdx.x * blockDim.x + threadIdx.x;
    if (e < e_cnt) unsafeAtomicAdd(&deg[dst[e]], 1.0f);
}

__global__ void k_rsqrt_inplace(float* deg, long long n) {
    long long i = (long long)blockIdx.x * blockDim.x + threadIdx.x;
    if (i < n) deg[i] = rsqrtf(deg[i]);   // deg >= 1 always (self loops)
}

__global__ void k_edge_norm(const long long* __restrict__ src, const long long* __restrict__ dst,
                            const float* __restrict__ dinv, float* __restrict__ norm, long long e_cnt) {
    long long e = (long long)blockIdx.x * blockDim.x + threadIdx.x;
    if (e < e_cnt) norm[e] = ONE_MINUS_ALPHA * dinv[src[e]] * dinv[dst[e]];
}

// ---------------- fused MLP via WMMA f32 16x16x4 ----------------
// 4 waves / block, each block computes 16 rows of h0 = relu(x@W1+b1)@W2+b2.

__global__ __launch_bounds__(128) void k_mlp_wmma(
    const float* __restrict__ x, const float* __restrict__ W1, const float* __restrict__ b1,
    const float* __restrict__ W2, const float* __restrict__ b2,
    float* __restrict__ h0_out, float* __restrict__ x0_copy, long long n_rows)
{
    __shared__ float xs[16 * IN_F];              // 32 KB: x tile, row-major
    __shared__ float w1p[HID_F * W1P_STRIDE];    // ~34 KB: W1 k-panel, [col][k] transposed
    __shared__ float Hs[16 * HID_F];             // 16 KB: hidden activations, row-major

    const int tid  = threadIdx.x;
    const int wave = tid >> 5;          // 0..3
    const int lane = tid & 31;
    const int m    = lane & 15;         // tile row (A) / tile col (B,C,D)
    const int kh   = lane >> 4;         // k-half select (0: K=0,1 ; 1: K=2,3)
    const long long row0 = (long long)blockIdx.x * 16;

    // Stage 0: load x tile (rows clamped for generic tail handling)
    for (int i = tid; i < 16 * IN_F / 4; i += 128) {
        int r = i / (IN_F / 4);
        int c = i % (IN_F / 4);
        long long gr = row0 + r; if (gr >= n_rows) gr = n_rows - 1;
        ((float4*)xs)[i] = ((const float4*)(x + gr * IN_F))[c];
    }
    __syncthreads();

    // Stage 1: H = relu(x@W1 + b1), wave `w` covers hidden cols [w*64, w*64+64)
    v8f acc[4];
    #pragma unroll
    for (int t = 0; t < 4; ++t)
        #pragma unroll
        for (int r = 0; r < 8; ++r) acc[t][r] = 0.0f;

    for (int k0 = 0; k0 < IN_F; k0 += 32) {
        __syncthreads();
        // load W1 rows [k0, k0+32) x 256 cols, transposed into w1p[col][k]
        for (int i = tid; i < 32 * HID_F / 4; i += 128) {
            int k  = i / (HID_F / 4);
            int c4 = (i % (HID_F / 4)) * 4;
            float4 v = ((const float4*)(W1 + (long long)(k0 + k) * HID_F))[c4 / 4];
            w1p[(c4 + 0) * W1P_STRIDE + k] = v.x;
            w1p[(c4 + 1) * W1P_STRIDE + k] = v.y;
            w1p[(c4 + 2) * W1P_STRIDE + k] = v.z;
            w1p[(c4 + 3) * W1P_STRIDE + k] = v.w;
        }
        __syncthreads();

        #pragma unroll
        for (int k = 0; k < 32; k += 4) {
            v2f a = *(const v2f*)&xs[m * IN_F + k0 + k + kh * 2];
            #pragma unroll
            for (int t = 0; t < 4; ++t) {
                int col = wave * 64 + t * 16 + m;
                v2f b = *(const v2f*)&w1p[col * W1P_STRIDE + k + kh * 2];
                acc[t] = __builtin_amdgcn_wmma_f32_16x16x4_f32(
                    false, a, false, b, (short)0, acc[t], false, false);
            }
        }
    }
    __syncthreads();

    // bias + relu -> LDS (C/D layout: lane<16 -> M=r, lane>=16 -> M=r+8)
    #pragma unroll
    for (int t = 0; t < 4; ++t) {
        int col = wave * 64 + t * 16 + m;
        float bb = b1[col];
        #pragma unroll
        for (int r = 0; r < 8; ++r) {
            float v = acc[t][r] + bb;
            Hs[(r + kh * 8) * HID_F + col] = v > 0.0f ? v : 0.0f;
        }
    }
    __syncthreads();

    // Stage 2: out = H@W2 + b2, wave `w` covers output cols [w*16, w*16+16)
    v8f acc2;
    #pragma unroll
    for (int r = 0; r < 8; ++r) acc2[r] = 0.0f;

    const int ocol = wave * 16 + m;
    #pragma unroll 4
    for (int k = 0; k < HID_F; k += 4) {
        v2f a = *(const v2f*)&Hs[m * HID_F + k + kh * 2];
        const float* bp = &W2[(k + kh * 2) * OUT_F + ocol];
        v2f b; b.x = bp[0]; b.y = bp[OUT_F];   // W2 is tiny & L2-hot
        acc2 = __builtin_amdgcn_wmma_f32_16x16x4_f32(
            false, a, false, b, (short)0, acc2, false, false);
    }

    float bb2 = b2[ocol];
    #pragma unroll
    for (int r = 0; r < 8; ++r) {
        long long gr = row0 + r + kh * 8;
        if (gr < n_rows) {
            float v = acc2[r] + bb2;
            h0_out[gr * OUT_F + ocol] = v;   // live h buffer (d_out)
            x0_copy[gr * OUT_F + ocol] = v;  // x0 for teleport term
        }
    }
}

// ---------------- APPNP propagation ----------------
// hnext = alpha*x0 + (1-alpha)*dinv[i]^2*hcur   (teleport + self-loop, float4)

__global__ __launch_bounds__(256) void k_step_init(
    const float* __restrict__ x0, const float* __restrict__ hcur,
    const float* __restrict__ dinv, float* __restrict__ hnext, long long n)
{
    long long i = (long long)blockIdx.x * blockDim.x + threadIdx.x;  // over n*16 float4s
    if (i >= n * (OUT_F / 4)) return;
    long long node = i >> 4;
    float di = dinv[node];
    float sw = ONE_MINUS_ALPHA * di * di;
    float4 c = ((const float4*)hcur)[i];
    float4 x = ((const float4*)x0)[i];
    float4 r;
    r.x = ALPHA_F * x.x + sw * c.x;
    r.y = ALPHA_F * x.y + sw * c.y;
    r.z = ALPHA_F * x.z + sw * c.z;
    r.w = ALPHA_F * x.w + sw * c.w;
    ((float4*)hnext)[i] = r;
}

// 16 lanes per edge: b128 gather of h[src], 4x global_atomic_add_f32 scatter to h[dst]
__global__ __launch_bounds__(256) void k_edge_prop(
    const long long* __restrict__ src, const long long* __restrict__ dst,
    const float* __restrict__ norm, const float* __restrict__ hcur,
    float* __restrict__ hnext, long long e_cnt)
{
    long long idx = (long long)blockIdx.x * blockDim.x + threadIdx.x;
    long long e = idx >> 4;
    if (e >= e_cnt) return;
    int l = threadIdx.x & 15;
    long long s = src[e];
    long long d = dst[e];
    float w = norm[e];
    float4 hv = *(const float4*)(hcur + s * OUT_F + l * 4);
    float* op = hnext + d * OUT_F + l * 4;
    unsafeAtomicAdd(op + 0, w * hv.x);
    unsafeAtomicAdd(op + 1, w * hv.y);
    unsafeAtomicAdd(op + 2, w * hv.z);
    unsafeAtomicAdd(op + 3, w * hv.w);
}

// ---------------- launch ----------------

extern "C" void kernel_launch(void* const* d_in, const int* in_sizes, int n_in,
                              void* d_out, int out_size, void* d_ws, size_t ws_size,
                              hipStream_t stream) {
    const float*     x  = (const float*)d_in[0];
    const long long* ei = (const long long*)d_in[1];   // int64 [2, E]
    const float*     W1 = (const float*)d_in[2];
    const float*     b1 = (const float*)d_in[3];
    const float*     W2 = (const float*)d_in[4];
    const float*     b2 = (const float*)d_in[5];
    float* out = (float*)d_out;

    const long long N = (long long)in_sizes[0] / IN_F;
    const long long E = (long long)in_sizes[1] / 2;
    const long long* src = ei;
    const long long* dst = ei + E;

    // workspace layout (all fp32)
    float* x0    = (float*)d_ws;          // N*64
    float* hping = x0 + N * OUT_F;        // N*64
    float* dinv  = hping + N * OUT_F;     // N   (deg -> rsqrt in place)
    float* norm  = dinv + N;              // E

    const int B = 256;
    dim3 blk(B);

    // degrees -> dinv -> edge norms
    k_init_deg<<<dim3((unsigned)((N + B - 1) / B)), blk, 0, stream>>>(dinv, N);
    k_deg_accum<<<dim3((unsigned)((E + B - 1) / B)), blk, 0, stream>>>(dst, dinv, E);
    k_rsqrt_inplace<<<dim3((unsigned)((N + B - 1) / B)), blk, 0, stream>>>(dinv, N);
    k_edge_norm<<<dim3((unsigned)((E + B - 1) / B)), blk, 0, stream>>>(src, dst, dinv, norm, E);

    // fused MLP -> h0 in d_out (live buffer), copy in x0
    k_mlp_wmma<<<dim3((unsigned)((N + 15) / 16)), dim3(128), 0, stream>>>(
        x, W1, b1, W2, b2, out, x0, N);

    // 10 propagation steps, ping-pong d_out <-> hping (even K ends in d_out)
    unsigned gi = (unsigned)((N * (OUT_F / 4) + B - 1) / B);
    unsigned ge = (unsigned)((E * 16 + B - 1) / B);
    for (int s = 0; s < 10; ++s) {
        const float* hc = (s % 2 == 0) ? out : hping;
        float*       hn = (s % 2 == 0) ? hping : out;
        k_step_init<<<dim3(gi), blk, 0, stream>>>(x0, hc, dinv, hn, N);
        k_edge_prop<<<dim3(ge), blk, 0, stream>>>(src, dst, norm, hc, hn, E);
    }
}